// GraphAstarModule_72980084294229
// MI455X (gfx1250) — compile-verified
//
#include <hip/hip_runtime.h>
#include <hip/hip_bf16.h>

typedef __attribute__((ext_vector_type(2))) float v2f;
typedef __attribute__((ext_vector_type(8))) float v8f;

// Problem constants (from reference)
#define BB 8
#define AA 6
#define NN 8192
#define SS 12
#define DD 8
#define TT 8
#define SD 96            // S*D
#define BA 48            // B*A
#define MTOT 393216      // B*A*N
#define LL 49152         // A*N
#define KSEL 1024        // k
#define KKEEP 8192       // K_keep
#define GAMMA 0.98f
#define FMASKED (-1.98e9f)   // gamma*(-1e9) + (-1e9)

// d_out float offsets
#define OFF_F  0
#define OFF_NV 3145728
#define OFF_TR 3538944
#define OFF_VS 3932160
#define OFF_SS 3997696
#define OFF_GS 10289152

#define LDA 100  // LDS row stride: 16B aligned rows, conflict-free frag reads

__device__ __forceinline__ unsigned orderKey(float x) {
    unsigned u = __float_as_uint(x);
    return (u & 0x80000000u) ? ~u : (u | 0x80000000u);
}

// ---------------------------------------------------------------------------
// Kernel 0: zero the top-k union mask (ws is poisoned between runs)
// ---------------------------------------------------------------------------
__global__ void zero_mask_kernel(int* __restrict__ m, int n) {
    int i = blockIdx.x * blockDim.x + threadIdx.x;
    if (i < n) m[i] = 0;
}

// ---------------------------------------------------------------------------
// Kernel 1: WMMA GEMM h = states @ W + b, then f = mask(h,g).
// One wave = one 16-row M tile; block = 8 waves = 128 rows.
// K=96 stepped by 4 with V_WMMA_F32_16X16X4_F32; B = W zero-padded to 16 cols.
// ---------------------------------------------------------------------------
__global__ __launch_bounds__(256) void gemm_f_kernel(
    const float* __restrict__ states, const unsigned char* __restrict__ valid,
    const float* __restrict__ g, const float* __restrict__ W,
    const float* __restrict__ bias, float* __restrict__ out_f) {
    __shared__ float sA[128 * LDA];
    __shared__ float sW[SD * 16];
    __shared__ float sB[16];

    const int tid = threadIdx.x;
    const int blockRow = blockIdx.x * 128;

    // Stage W (96x8 -> 96x16 zero-padded) and bias into LDS
    for (int i = tid; i < SD * 16; i += 256) {
        int r = i >> 4, c = i & 15;
        sW[i] = (c < TT) ? W[r * TT + c] : 0.0f;
    }
    if (tid < 16) sB[tid] = (tid < TT) ? bias[tid] : 0.0f;

    // Stage A tile (128 rows x 96 floats) coalesced as float4
    const float4* src = (const float4*)(states + (size_t)blockRow * SD);
    for (int i = tid; i < 128 * 24; i += 256) {
        int r = i / 24, c = i % 24;
        float4 v = src[(size_t)r * 24 + c];
        float* dst = &sA[r * LDA + c * 4];
        dst[0] = v.x; dst[1] = v.y; dst[2] = v.z; dst[3] = v.w;
    }
    __syncthreads();

    const int wave = tid >> 5;
    const int lane = tid & 31;
    const int hi = lane >> 4;      // half-wave select (K +0/+2 slot)
    const int lm = lane & 15;
    const int rowL = wave * 16 + lm;

    v8f acc = {};
    for (int k4 = 0; k4 < SD; k4 += 4) {
        // A frag: lanes 0-15 hold (M=lm, K=k4+{0,1}); lanes 16-31 (M=lm, K=k4+{2,3})
        v2f a, bf;
        int ca = k4 + hi * 2;
        a.x = sA[rowL * LDA + ca];
        a.y = sA[rowL * LDA + ca + 1];
        // B frag mirrors: row K across half-waves, N = lm
        bf.x = sW[(k4 + hi * 2) * 16 + lm];
        bf.y = sW[(k4 + hi * 2 + 1) * 16 + lm];
        acc = __builtin_amdgcn_wmma_f32_16x16x4_f32(
            /*neg_a=*/false, a, /*neg_b=*/false, bf,
            /*c_mod=*/(short)0, acc, /*reuse_a=*/false, /*reuse_b=*/false);
    }

    // D layout: VGPR r -> (M=r, N=lane) lanes 0-15 ; (M=r+8, N=lane-16) lanes 16-31
    float bcol = sB[lm];
    if (lm < TT) {
        int t = lm;
        #pragma unroll
        for (int r = 0; r < 8; ++r) {
            int mrow = blockRow + wave * 16 + r + hi * 8;
            float h = acc[r] + bcol;
            int vb = valid[mrow];
            float gv = g[(size_t)mrow * TT + t];
            out_f[(size_t)mrow * TT + t] = vb ? (GAMMA * h + gv) : FMASKED;
        }
    }
}

// ---------------------------------------------------------------------------
// Kernel 2: exact top-k (k=1024) per (ba, t) column via 4-pass radix select
// on monotone keys; ties taken in ascending index order; union via atomicOr.
// grid = (48, 8), block = 256.
// ---------------------------------------------------------------------------
__global__ __launch_bounds__(256) void topk_select_kernel(
    const float* __restrict__ f, int* __restrict__ topkMask, int k) {
    const int ba = blockIdx.x;
    const int t = blockIdx.y;
    const float* fp = f + (size_t)ba * NN * TT + t;   // element n at fp[n*TT]

    __shared__ int hist[256];
    __shared__ int scn[256];
    __shared__ unsigned s_prefix;
    __shared__ int s_want;

    const int tid = threadIdx.x;
    if (tid == 0) { s_prefix = 0u; s_want = k; }
    __syncthreads();

    for (int pass = 0; pass < 4; ++pass) {
        const int shift = 24 - pass * 8;
        const unsigned pmask = (pass == 0) ? 0u : (0xFFFFFFFFu << (shift + 8));
        hist[tid] = 0;
        __syncthreads();
        const unsigned prefix = s_prefix;
        for (int i = tid; i < NN; i += 256) {
            unsigned key = orderKey(fp[(size_t)i * TT]);
            if ((key & pmask) == prefix)
                atomicAdd(&hist[(key >> shift) & 255], 1);
        }
        __syncthreads();
        if (tid == 0) {
            int want0 = s_want, cum = 0;
            for (int d = 255; d >= 0; --d) {
                int h = hist[d];
                if (cum + h >= want0 || d == 0) {
                    s_prefix = prefix | ((unsigned)d << shift);
                    s_want = want0 - cum;
                    break;
                }
                cum += h;
            }
        }
        __syncthreads();
    }

    const unsigned thr = s_prefix;
    const int rem = s_want;   // how many threshold-equal keys to keep

    // rank threshold-equal elements in index order: contiguous chunks + scan
    const int CH = NN / 256;  // 32
    const int n0 = tid * CH;
    int eq = 0;
    for (int i = 0; i < CH; ++i)
        if (orderKey(fp[(size_t)(n0 + i) * TT]) == thr) eq++;
    scn[tid] = eq;
    __syncthreads();
    for (int off = 1; off < 256; off <<= 1) {
        int add = (tid >= off) ? scn[tid - off] : 0;
        __syncthreads();
        scn[tid] += add;
        __syncthreads();
    }
    int running = scn[tid] - eq;  // exclusive prefix of equals
    for (int i = 0; i < CH; ++i) {
        int n = n0 + i;
        unsigned key = orderKey(fp[(size_t)n * TT]);
        bool take = false;
        if (key > thr) take = true;
        else if (key == thr) { if (running < rem) take = true; running++; }
        if (take) atomicOr(&topkMask[ba * NN + n], 1);
    }
}

// ---------------------------------------------------------------------------
// Kernel 3: new_valid = valid & topk
// ---------------------------------------------------------------------------
__global__ void combine_kernel(const unsigned char* __restrict__ valid,
                               const int* __restrict__ topkMask,
                               int* __restrict__ nvMask,
                               float* __restrict__ out_nv, int n) {
    int i = blockIdx.x * blockDim.x + threadIdx.x;
    if (i >= n) return;
    int nv = (valid[i] != 0) && (topkMask[i] != 0);
    nvMask[i] = nv;
    out_nv[i] = (float)nv;
}

// ---------------------------------------------------------------------------
// Kernel 4: per-batch inclusive scan over A*N=49152 -> transition + cumsum.
// One block (1024 threads, 48 elems each) per batch.
// ---------------------------------------------------------------------------
__global__ __launch_bounds__(1024) void scan_transition_kernel(
    const int* __restrict__ nvMask, int* __restrict__ cums,
    float* __restrict__ out_tr) {
    const int b = blockIdx.x;
    const int CH = LL / 1024;  // 48
    const int* m = nvMask + (size_t)b * LL;
    const int tid = threadIdx.x;
    __shared__ int ssum[1024];

    const int base = tid * CH;
    int s = 0;
    for (int i = 0; i < CH; ++i) s += m[base + i];
    ssum[tid] = s;
    __syncthreads();
    for (int off = 1; off < 1024; off <<= 1) {
        int add = (tid >= off) ? ssum[tid - off] : 0;
        __syncthreads();
        ssum[tid] += add;
        __syncthreads();
    }
    int run = ssum[tid] - s;  // exclusive prefix for this chunk
    for (int i = 0; i < CH; ++i) {
        int j = base + i;
        int v = m[j];
        run += v;
        cums[(size_t)b * LL + j] = run;
        out_tr[(size_t)b * LL + j] = v ? (float)(run - 1) : -1.0f;
    }
}

// ---------------------------------------------------------------------------
// Kernel 5: stream compaction. Valid entries first (rank = cum-1), then
// invalid entries in ascending index order — matches top_k-on-mask gather.
// ---------------------------------------------------------------------------
__global__ void compact_kernel(const float* __restrict__ states,
                               const float* __restrict__ g,
                               const int* __restrict__ nvMask,
                               const int* __restrict__ cums,
                               float* __restrict__ out_vs,
                               float* __restrict__ out_ss,
                               float* __restrict__ out_gs) {
    int gid = blockIdx.x * blockDim.x + threadIdx.x;
    if (gid >= BB * LL) return;
    int b = gid / LL, j = gid % LL;
    const int* cumb = cums + (size_t)b * LL;
    int total = cumb[LL - 1];
    int inc = cumb[j];
    int v = nvMask[gid];
    int p = v ? (inc - 1) : (total + (j - inc));
    if (p >= KKEEP) return;

    out_vs[(size_t)b * KKEEP + p] = (float)v;
    const float* gs = g + (size_t)gid * TT;
    float* gd = out_gs + ((size_t)b * KKEEP + p) * TT;
    #pragma unroll
    for (int i = 0; i < TT; ++i) gd[i] = gs[i];
    const float4* ss = (const float4*)(states + (size_t)gid * SD);
    float4* sd = (float4*)(out_ss + ((size_t)b * KKEEP + p) * SD);
    #pragma unroll
    for (int i = 0; i < SD / 4; ++i) sd[i] = ss[i];
}

// ---------------------------------------------------------------------------
extern "C" void kernel_launch(void* const* d_in, const int* in_sizes, int n_in,
                              void* d_out, int out_size, void* d_ws, size_t ws_size,
                              hipStream_t stream) {
    const float* states        = (const float*)d_in[0];
    const unsigned char* valid = (const unsigned char*)d_in[1];
    const float* g             = (const float*)d_in[2];
    const float* W             = (const float*)d_in[3];
    const float* bias          = (const float*)d_in[4];
    // d_in[5] = k (1024), d_in[6] = K_keep (8192): static in reference.

    float* out    = (float*)d_out;
    float* out_f  = out + OFF_F;
    float* out_nv = out + OFF_NV;
    float* out_tr = out + OFF_TR;
    float* out_vs = out + OFF_VS;
    float* out_ss = out + OFF_SS;
    float* out_gs = out + OFF_GS;

    char* ws = (char*)d_ws;
    int* topkMask = (int*)(ws);
    int* nvMask   = (int*)(ws + (size_t)MTOT * 4);
    int* cums     = (int*)(ws + (size_t)MTOT * 8);

    zero_mask_kernel<<<(MTOT + 255) / 256, 256, 0, stream>>>(topkMask, MTOT);
    gemm_f_kernel<<<MTOT / 128, 256, 0, stream>>>(states, valid, g, W, bias, out_f);
    dim3 gsel(BA, TT);
    topk_select_kernel<<<gsel, 256, 0, stream>>>(out_f, topkMask, KSEL);
    combine_kernel<<<(MTOT + 255) / 256, 256, 0, stream>>>(valid, topkMask, nvMask, out_nv, MTOT);
    scan_transition_kernel<<<BB, 1024, 0, stream>>>(nvMask, cums, out_tr);
    compact_kernel<<<(BB * LL + 255) / 256, 256, 0, stream>>>(states, g, nvMask, cums,
                                                              out_vs, out_ss, out_gs);
}